// CurveGraphic2d_11287174054429
// MI455X (gfx1250) — compile-verified
//
#include <hip/hip_runtime.h>
#include <math.h>

typedef __attribute__((ext_vector_type(2))) float v2f;
typedef __attribute__((ext_vector_type(8))) float v8f;

#define HDIM   256
#define WDIM   256
#define NSAMP  32
#define NCURVE 16

// Fast-path device math: single-instruction TRANS ops on gfx1250.
__device__ __forceinline__ float fast_sqrt(float x) { return __builtin_amdgcn_sqrtf(x); }
__device__ __forceinline__ float fast_log2(float x) { return __builtin_amdgcn_logf(x); }   // v_log_f32 = log2
__device__ __forceinline__ float fast_exp2(float x) { return __builtin_amdgcn_exp2f(x); }  // v_exp_f32 = 2^x

// ---------------------------------------------------------------------------
// Kernel 1: per-curve setup (arc length -> truncation -> de Casteljau -> 32
// sample points). 16 curves, one thread each; trivially cheap.
// ws layout: float2 sample[b][s] = (y, x), b in [0,16), s in [0,32)
// ---------------------------------------------------------------------------
__device__ __forceinline__ void bez4(const float* py, const float* px, float t,
                                     float& oy, float& ox) {
  float u  = 1.0f - t;
  float b0 = u * u * u;
  float b1 = 3.0f * t * u * u;
  float b2 = 3.0f * t * t * u;
  float b3 = t * t * t;
  oy = b0 * py[0] + b1 * py[1] + b2 * py[2] + b3 * py[3];
  ox = b0 * px[0] + b1 * px[1] + b2 * px[2] + b3 * px[3];
}

__global__ void curve_setup(const float* __restrict__ in, float* __restrict__ ws) {
  int b = threadIdx.x;
  if (b >= NCURVE) return;

  float cy[4], cx[4];
#pragma unroll
  for (int k = 0; k < 4; ++k) {
    cy[k] = in[b * 8 + k * 2 + 0] * (float)HDIM;
    cx[k] = in[b * 8 + k * 2 + 1] * (float)WDIM;
  }

  // Arc length of 32-sample polyline on the original curve.
  float py, px;
  bez4(cy, cx, 0.0f, py, px);
  float arc = 0.0f;
  for (int s = 1; s < NSAMP; ++s) {
    float t = (float)s * (1.0f / 31.0f);
    float qy, qx;
    bez4(cy, cx, t, qy, qx);
    float dy = qy - py, dx = qx - px;
    arc += fast_sqrt(dy * dy + dx * dx);
    py = qy; px = qx;
  }
  float tt = fminf(1.0f, 300.0f / (arc + 1e-6f));

  // Left de Casteljau subdivision at tt.
  float qy[4], qx[4], ly[4], lx[4];
#pragma unroll
  for (int k = 0; k < 4; ++k) { qy[k] = cy[k]; qx[k] = cx[k]; }
  ly[0] = qy[0]; lx[0] = qx[0];
#pragma unroll
  for (int lvl = 1; lvl < 4; ++lvl) {
#pragma unroll
    for (int i = 0; i + lvl < 4; ++i) {
      qy[i] = (1.0f - tt) * qy[i] + tt * qy[i + 1];
      qx[i] = (1.0f - tt) * qx[i] + tt * qx[i + 1];
    }
    ly[lvl] = qy[0]; lx[lvl] = qx[0];
  }

  // Resample truncated curve at 32 points.
  float2* sp = (float2*)ws + b * NSAMP;
  for (int s = 0; s < NSAMP; ++s) {
    float t = (float)s * (1.0f / 31.0f);
    float oy, ox;
    bez4(ly, lx, t, oy, ox);
    sp[s] = make_float2(oy, ox);
  }
}

// ---------------------------------------------------------------------------
// Kernel 2: WMMA rasterizer.
//   D[m,n] = -2 s_m . p_n + ||s_m||^2   via  V_WMMA_F32_16X16X4_F32
//   A (16 samples x 4): [-2sy, -2sx, ||s||^2, 0]   (row-centered coords)
//   B (4 x 16 pixels) : [py_rel(=0), px_rel, 1, 0]
//   min over samples = 15 lane-local v_min + one shfl_xor(16).
// One wave <-> one (curve,row); 16 tiles of 16 pixels per wave.
// Finalize uses fast-path transcendentals: base > 0 always, so
// pow(base, aa) == exp2(aa * log2(base)) in the positive domain.
// ---------------------------------------------------------------------------
__global__ __launch_bounds__(256) void curve_render(
    const float* __restrict__ ws, const float* __restrict__ widths,
    const float* __restrict__ aaf, float* __restrict__ out) {
  const int lane = threadIdx.x & 31;
  const int wv   = threadIdx.x >> 5;
  const int row  = blockIdx.x * 8 + wv;   // canvas row
  const int b    = blockIdx.y;            // curve
  const int sl   = lane & 15;
  const bool hi  = lane >= 16;

  const float2* sp = (const float2*)ws + b * NSAMP;
  float2 slo = sp[sl];        // sample sl       (rows of WMMA #0)
  float2 shi = sp[sl + 16];   // sample sl + 16  (rows of WMMA #1)

  // Row-centered coordinates: cy = row (so py_rel == 0), cx = 127.5.
  const float cy = (float)row;
  const float cx = 127.5f;
  float dyl = slo.x - cy, dxl = slo.y - cx;
  float dyh = shi.x - cy, dxh = shi.y - cx;

  // A-matrix VGPR layout (16x4 f32): lanes 0-15 hold K=0,1; lanes 16-31 hold K=2,3.
  v2f aLo, aHi, bv;
  aLo.x = hi ? (dyl * dyl + dxl * dxl) : (-2.0f * dyl);
  aLo.y = hi ? 0.0f                    : (-2.0f * dxl);
  aHi.x = hi ? (dyh * dyh + dxh * dxh) : (-2.0f * dyh);
  aHi.y = hi ? 0.0f                    : (-2.0f * dxh);
  // B-matrix (4x16 f32): lanes 0-15 hold K=0,1 of column n; lanes 16-31 hold K=2,3.
  bv.x = hi ? 1.0f : 0.0f;   // K=0 row: py_rel = 0 ; K=2 row: constant 1

  const float invw = 1.0f / widths[b];
  const float aa   = aaf[b];
  float* orow = out + ((size_t)b * HDIM + row) * WDIM;

  v8f zero = {};
#pragma unroll 4
  for (int tile = 0; tile < 16; ++tile) {
    float pxr = (float)(tile * 16 + sl) - 127.5f;  // pixel x, row-centered
    bv.y = hi ? 0.0f : pxr;                        // K=1 row: px_rel ; K=3 row: 0

    v8f d0 = __builtin_amdgcn_wmma_f32_16x16x4_f32(
        false, aLo, false, bv, (short)0, zero, false, false);
    v8f d1 = __builtin_amdgcn_wmma_f32_16x16x4_f32(
        false, aHi, false, bv, (short)0, zero, false, false);

    float m = fminf(d0[0], d1[0]);
#pragma unroll
    for (int i = 1; i < 8; ++i) m = fminf(m, fminf(d0[i], d1[i]));
    // lane l holds samples {0..7,16..23}, lane l^16 holds {8..15,24..31} of the same pixel.
    m = fminf(m, __shfl_xor(m, 16, 32));

    float d2   = fmaxf(m + pxr * pxr, 0.0f);          // + ||p_rel||^2
    float dmin = fast_sqrt(d2);                       // single v_sqrt_f32
    float base = dmin * invw + 1e-6f;                 // > 0 always
    float v    = 1.0f - fast_exp2(aa * fast_log2(base)); // == powf(base, aa)
    v = fminf(fmaxf(v, 0.0f), 1.0f);
    if (!hi) orow[tile * 16 + sl] = v;
  }
}

// ---------------------------------------------------------------------------
extern "C" void kernel_launch(void* const* d_in, const int* in_sizes, int n_in,
                              void* d_out, int out_size, void* d_ws, size_t ws_size,
                              hipStream_t stream) {
  const float* inputs = (const float*)d_in[0];   // [16,4,2] normalized keypoints
  const float* widths = (const float*)d_in[1];   // [16]
  const float* aaf    = (const float*)d_in[2];   // [16]
  float*       out    = (float*)d_out;           // [16,256,256]
  float*       ws     = (float*)d_ws;            // 16*32*2 floats used

  curve_setup<<<dim3(1), dim3(32), 0, stream>>>(inputs, ws);
  curve_render<<<dim3(HDIM / 8, NCURVE), dim3(256), 0, stream>>>(ws, widths, aaf, out);

  (void)in_sizes; (void)n_in; (void)out_size; (void)ws_size;
}